// MaskRCNNBoxProposal_18897856102508
// MI455X (gfx1250) — compile-verified
//
#include <hip/hip_runtime.h>
#include <stdint.h>

#define IMG_F 800.0f
#define NANCH 159882
#define NSEL 4507
#define KEYPAD 8192
#define POSTN 1000
#define NMS_TH 0.7f
#define BBOX_CLIP_F 4.135166556742356f

typedef float v2f  __attribute__((ext_vector_type(2)));
typedef float v8f  __attribute__((ext_vector_type(8)));
typedef unsigned int u32x4 __attribute__((ext_vector_type(4)));
typedef int   i32x4 __attribute__((ext_vector_type(4)));
typedef int   i32x8 __attribute__((ext_vector_type(8)));

#if defined(__has_builtin)
# if __has_builtin(__builtin_amdgcn_tensor_load_to_lds) && __has_builtin(__builtin_amdgcn_s_wait_tensorcnt)
#  define HAVE_TDM 1
# endif
# if __has_builtin(__builtin_amdgcn_wmma_f32_16x16x4_f32)
#  define HAVE_WMMA 1
# endif
#endif
#ifndef HAVE_TDM
# define HAVE_TDM 0
#endif
#ifndef HAVE_WMMA
# define HAVE_WMMA 0
#endif

__device__ __constant__ int   c_ln[5]  = {120000,30000,7500,1875,507};
__device__ __constant__ int   c_lo[5]  = {0,120000,150000,157500,159375};
__device__ __constant__ int   c_gw[5]  = {200,100,50,25,13};
__device__ __constant__ int   c_st[5]  = {4,8,16,32,61};
__device__ __constant__ float c_sz[5]  = {32.f,64.f,128.f,256.f,512.f};
__device__ __constant__ int   c_sk[5]  = {1000,1000,1000,1000,507};
__device__ __constant__ int   c_so[5]  = {0,1000,2000,3000,4000};
__device__ __constant__ float c_scale[4] = {0.25f,0.125f,0.0625f,0.03125f};
__device__ __constant__ int   c_fhw[4] = {200,100,50,25};

struct InPtrs   { const float* obj[5]; const float* del[5]; };
struct FeatPtrs { const float* f[4];   const float* ft[4];  };

__device__ inline unsigned flipf(float f){
  unsigned u = __float_as_uint(f);
  return (u & 0x80000000u) ? ~u : (u | 0x80000000u);
}

// ---------------------------------------------------------------- K1: decode
__global__ void k_decode(InPtrs in, float* props, float* scr, float* maxc, int B){
  int gid = blockIdx.x*blockDim.x + threadIdx.x;
  if (gid < B) ((unsigned*)maxc)[gid] = 0u;      // per-launch init for atomicMax
  int total = B * NANCH;
  if (gid >= total) return;
  int b = gid / NANCH, i = gid - b*NANCH;
  int l = 0;
  while (l < 4 && i >= c_lo[l+1]) ++l;
  int j = i - c_lo[l];
  int cell = j/3, a = j - cell*3;
  int gw = c_gw[l];
  int h = cell / gw, w2 = cell - h*gw;
  float xx = (float)(w2 * c_st[l]), yy = (float)(h * c_st[l]);
  float ratio = (a==0)?0.5f:((a==1)?1.0f:2.0f);
  float hr = sqrtf(ratio);
  float wsz = c_sz[l] / hr, hsz = c_sz[l]*hr;
  float ax1 = xx + rintf(-0.5f*wsz), ay1 = yy + rintf(-0.5f*hsz);
  float ax2 = xx + rintf( 0.5f*wsz), ay2 = yy + rintf( 0.5f*hsz);
  float wa = ax2-ax1, ha = ay2-ay1;
  float cxa = ax1 + 0.5f*wa, cya = ay1 + 0.5f*ha;
  int hw = gw*gw;
  int pix = h*gw + w2;
  const float* op = in.obj[l]; const float* dp = in.del[l];
  float o  = op[((size_t)b*3 + a)*hw + pix];
  float dx = dp[((size_t)b*12 + a*4 + 0)*hw + pix];
  float dy = dp[((size_t)b*12 + a*4 + 1)*hw + pix];
  float dw = fminf(dp[((size_t)b*12 + a*4 + 2)*hw + pix], BBOX_CLIP_F);
  float dh = fminf(dp[((size_t)b*12 + a*4 + 3)*hw + pix], BBOX_CLIP_F);
  float cx = dx*wa + cxa, cy = dy*ha + cya;
  float ww = expf(dw)*wa, hh = expf(dh)*ha;
  float* P = props + ((size_t)b*NANCH + i)*4;
  P[0]=cx-0.5f*ww; P[1]=cy-0.5f*hh; P[2]=cx+0.5f*ww; P[3]=cy+0.5f*hh;
  scr[(size_t)b*NANCH + i] = o;
}

// ------------------------------------------- K2: radix-select kth threshold
__global__ void k_thresh(const float* scr, unsigned* thrT, int* needE, int B){
  int bl = blockIdx.x; int b = bl/5, l = bl - b*5;
  int n = c_ln[l], off = c_lo[l], k = c_sk[l];
  const float* d = scr + (size_t)b*NANCH + off;
  __shared__ unsigned hist[256];
  __shared__ unsigned sh_prefix;
  __shared__ int sh_k;
  __shared__ unsigned sh_g;
  int tid = threadIdx.x;
  if (tid==0){ sh_prefix=0; sh_k=k; sh_g=0; }
  __syncthreads();
  for (int byte=3; byte>=0; --byte){
    for (int i=tid;i<256;i+=blockDim.x) hist[i]=0;
    __syncthreads();
    unsigned pfx = sh_prefix;
    for (int i=tid;i<n;i+=blockDim.x){
      unsigned key = flipf(d[i]);
      bool match = (byte==3) || ((key >> ((byte+1)*8)) == (pfx >> ((byte+1)*8)));
      if (match) atomicAdd(&hist[(key >> (byte*8)) & 255], 1u);
    }
    __syncthreads();
    if (tid==0){
      unsigned cum=0; int kk = sh_k;
      for (int bin=255; bin>=0; --bin){
        unsigned c = hist[bin];
        if (cum + c >= (unsigned)kk){
          sh_prefix = pfx | ((unsigned)bin << (byte*8));
          sh_k = kk - (int)cum;
          break;
        }
        cum += c;
      }
    }
    __syncthreads();
  }
  unsigned T = sh_prefix;
  unsigned g = 0;
  for (int i=tid;i<n;i+=blockDim.x) if (flipf(d[i]) > T) ++g;
  atomicAdd(&sh_g, g);
  __syncthreads();
  if (tid==0){ thrT[bl] = T; needE[bl] = k - (int)sh_g; }
}

// -------------------------- K3: deterministic compaction (index-order ties)
__global__ void k_compact(const float* scr, const unsigned* thrT, const int* needE,
                          int* sel, int B){
  int bl = blockIdx.x; int b = bl/5, l = bl - b*5;
  int n = c_ln[l], off = c_lo[l], k = c_sk[l];
  unsigned T = thrT[bl]; int ne = needE[bl]; int G = k - ne;
  const float* d = scr + (size_t)b*NANCH + off;
  int* out = sel + (size_t)b*NSEL + c_so[l];
  __shared__ int sc[256];
  __shared__ int baseg, basee;
  int tid = threadIdx.x;
  if (tid==0){ baseg=0; basee=0; }
  __syncthreads();
  for (int base=0; base<n; base+=256){
    int i = base + tid;
    unsigned key = (i<n) ? flipf(d[i]) : 0u;
    int pg = (i<n && key >  T) ? 1 : 0;
    int pe = (i<n && key == T) ? 1 : 0;
    sc[tid]=pg; __syncthreads();
    for (int dd=1; dd<256; dd<<=1){ int t=(tid>=dd)?sc[tid-dd]:0; __syncthreads(); sc[tid]+=t; __syncthreads(); }
    int incg = sc[tid]; int tg = sc[255];
    if (pg) out[baseg + incg - 1] = off + i;
    __syncthreads();
    sc[tid]=pe; __syncthreads();
    for (int dd=1; dd<256; dd<<=1){ int t=(tid>=dd)?sc[tid-dd]:0; __syncthreads(); sc[tid]+=t; __syncthreads(); }
    int ince = sc[tid]; int te = sc[255];
    if (pe){ int pos = basee + ince - 1; if (pos < ne) out[G + pos] = off + i; }
    __syncthreads();
    if (tid==0){ baseg += tg; basee += te; }
    __syncthreads();
  }
}

// --------------------- K4: gather selected, clip, sigmoid, pack sort keys
__global__ void k_pack(const float* props, const float* scr, const int* sel,
                       float* bsel, int* vsel, unsigned long long* keys,
                       float* maxc, int B){
  int b = blockIdx.x >> 3;
  int slot = ((blockIdx.x & 7) << 10) + threadIdx.x;
  if (slot >= KEYPAD) return;
  unsigned long long* kk = keys + (size_t)b*KEYPAD;
  if (slot >= NSEL){ kk[slot] = 0ull; return; }
  int idx = sel[(size_t)b*NSEL + slot];
  const float* P = props + ((size_t)b*NANCH + idx)*4;
  float x1 = fminf(fmaxf(P[0],0.f),IMG_F);
  float y1 = fminf(fmaxf(P[1],0.f),IMG_F);
  float x2 = fminf(fmaxf(P[2],0.f),IMG_F);
  float y2 = fminf(fmaxf(P[3],0.f),IMG_F);
  int valid = (x2-x1 >= 0.001f) && (y2-y1 >= 0.001f);
  float o = scr[(size_t)b*NANCH + idx];
  float s = 1.0f/(1.0f + expf(-o));
  float sm = valid ? s : __uint_as_float(0xFF800000u);  // -inf when invalid
  float* Q = bsel + ((size_t)b*NSEL + slot)*4;
  Q[0]=x1; Q[1]=y1; Q[2]=x2; Q[3]=y2;
  vsel[(size_t)b*NSEL + slot] = valid;
  float m4 = fmaxf(fmaxf(x1,y1), fmaxf(x2,y2));
  atomicMax((unsigned*)&maxc[b], __float_as_uint(m4));   // all coords >= 0
  // descending key; smaller original slot wins ties (stable like jnp.argsort)
  kk[slot] = ((unsigned long long)flipf(sm) << 32)
           | (unsigned long long)(0xFFFFFFFFu - (unsigned)slot);
}

// ------------------ K5: per-image bitonic sort (TDM stages keys into LDS)
__global__ __launch_bounds__(1024)
void k_sort(const unsigned long long* keys, int* order, int B){
  __shared__ unsigned long long sk[KEYPAD];   // 64 KB
  int b = blockIdx.x, tid = threadIdx.x;
  const unsigned long long* src = keys + (size_t)b*KEYPAD;
#if HAVE_TDM
  if (tid < 32){
    // 1D TDM copy: 16384 dwords (64 KB) global -> LDS.  D# per CDNA5 ISA ch.8.
    unsigned lds_off = (unsigned)(uintptr_t)&sk[0];       // LDS aperture: low 32 bits = offset
    unsigned long long ga = (unsigned long long)(uintptr_t)src;
    u32x4 g0 = { 1u,                                      // count=1 (valid descriptor)
                 lds_off,
                 (unsigned)(ga & 0xFFFFFFFFull),
                 (unsigned)((ga >> 32) & 0x01FFFFFFull) | 0x80000000u }; // type=2
    i32x8 g1 = { 0x00020000,          // data_size=4B, no multicast/pad/iterate
                 (int)0x40000000,     // tensor_dim0 = 16384 (bits 79:48 low half)
                 0,                   // tensor_dim0 hi / tensor_dim1 lo
                 (int)0x40000000,     // tile_dim0 = 16384 (bits 127:112)
                 0, 16384, 0, 0 };    // tile_dim1/2=0, tensor_dim0_stride=16384
    i32x4 z4 = {0,0,0,0};
#if __clang_major__ >= 23
    i32x8 z8 = {0,0,0,0,0,0,0,0};
    __builtin_amdgcn_tensor_load_to_lds(g0, g1, z4, z4, z8, 0);
#else
    __builtin_amdgcn_tensor_load_to_lds(g0, g1, z4, z4, 0);
#endif
    __builtin_amdgcn_s_wait_tensorcnt((short)0);
  }
#endif
  __syncthreads();
  // TDM semantics are not hardware-verified: overwrite with a normal copy so
  // the result is correct either way (same data, same destination).
  for (int i=tid;i<KEYPAD;i+=1024) sk[i]=src[i];
  __syncthreads();
  for (unsigned kk=2; kk<=KEYPAD; kk<<=1){
    for (unsigned j=kk>>1; j>0; j>>=1){
      for (unsigned i=tid; i<KEYPAD; i+=1024){
        unsigned ixj = i^j;
        if (ixj > i){
          unsigned long long a = sk[i], c = sk[ixj];
          bool up = ((i & kk) == 0);
          bool sw = up ? (a < c) : (a > c);   // descending sort
          if (sw){ sk[i]=c; sk[ixj]=a; }
        }
      }
      __syncthreads();
    }
  }
  for (int r=tid;r<NSEL;r+=1024)
    order[(size_t)b*NSEL + r] = (int)(0xFFFFFFFFu - (unsigned)(sk[r] & 0xFFFFFFFFull));
}

// --------------------------------------------- K6: serial greedy NMS
__global__ __launch_bounds__(1024)
void k_nms(const float* bsel, const int* vsel, const int* order,
           const float* maxc, float* out_final, int B){
  int b = blockIdx.x, tid = threadIdx.x;
  __shared__ unsigned char supp[NSEL];
  __shared__ unsigned char vld[NSEL];
  __shared__ int oslot[NSEL];
  __shared__ int sh_ki;
  __shared__ float sb0,sb1,sb2,sb3,sba;
  float mc = maxc[b];
  float ox1[5],oy1[5],ox2[5],oy2[5],oar[5];
  float q0r[5],q1r[5],q2r[5],q3r[5];
  for (int m=0;m<5;m++){
    int j = tid + (m<<10);
    if (j < NSEL){
      int o = order[(size_t)b*NSEL + j];
      const float* Q = bsel + ((size_t)b*NSEL + o)*4;
      float q0=Q[0],q1=Q[1],q2=Q[2],q3=Q[3];
      q0r[m]=q0;q1r[m]=q1;q2r[m]=q2;q3r[m]=q3;
      float offv = (float)(o/1000) * (mc + 1.0f);   // level offset
      ox1[m]=q0+offv; oy1[m]=q1+offv; ox2[m]=q2+offv; oy2[m]=q3+offv;
      oar[m]=(q2-q0)*(q3-q1);
      supp[j]=0;
      vld[j]=(unsigned char)vsel[(size_t)b*NSEL + o];
    }
  }
  for (int i=tid;i<POSTN*4;i+=1024) out_final[(size_t)b*POSTN*4 + i] = 0.0f;
  __syncthreads();
  int nk = 0;
  for (int i=0;i<NSEL;i++){
    if (tid == (i & 1023)){
      int m = i >> 10;
      sb0=ox1[m]; sb1=oy1[m]; sb2=ox2[m]; sb3=oy2[m]; sba=oar[m];
    }
    if (tid == 0){
      int ki = vld[i] && !supp[i];
      sh_ki = ki;
      oslot[i] = (ki && nk < POSTN) ? nk : -1;
      if (ki) nk++;
    }
    __syncthreads();
    if (sh_ki){
      for (int m=0;m<5;m++){
        int j = tid + (m<<10);
        if (j < NSEL){
          float lx = fmaxf(sb0, ox1[m]), ly = fmaxf(sb1, oy1[m]);
          float rx = fminf(sb2, ox2[m]), ry = fminf(sb3, oy2[m]);
          float iw = fmaxf(rx-lx, 0.f), ih = fmaxf(ry-ly, 0.f);
          float inter = iw*ih;
          float iou = inter / (sba + oar[m] - inter);
          if (iou > NMS_TH) supp[j]=1;
        }
      }
    }
    __syncthreads();
  }
  for (int m=0;m<5;m++){
    int j = tid + (m<<10);
    if (j < NSEL){
      int s = oslot[j];
      if (s >= 0){
        float* O = out_final + ((size_t)b*POSTN + s)*4;
        O[0]=q0r[m];O[1]=q1r[m];O[2]=q2r[m];O[3]=q3r[m];
      }
    }
  }
}

// -------------------- K7: feature transpose to channels-last (coalesced gather)
__global__ void k_transpose(const float* f, float* ft, int B, int H, int W){
  int gid = blockIdx.x*blockDim.x + threadIdx.x;
  int total = B*H*W*256;
  if (gid >= total) return;
  int c = gid & 255;
  int rest = gid >> 8;
  int x = rest % W; int rest2 = rest / W;
  int y = rest2 % H; int b = rest2 / H;
  ft[gid] = f[(((size_t)(b*256 + c))*H + y)*W + x];
}

// --------------------- K8: ROI-Align as WMMA:  OUT[49,16] = W[49,784]·G[784,16]
// Each K-point p=(t,u) is owned by exactly one output cell (t>>2)*7+(u>>2), so
// the dense W matrix is materialized as a per-ROI LDS table {weight, owner}
// and A-fragments are built branch-free with compare+select (no EXEC churn).
__global__ __launch_bounds__(128)
void k_roialign(FeatPtrs fp, const float* final_boxes, float* out_feats,
                int B, int use_t){
  __shared__ float Gs[784][16];           // 50 KB: gathered 28x28 corner grid
  __shared__ float awp[784][2];           // 6.1 KB: {ayw[t]*axw[u], owner cell}
  __shared__ int   xcor[28], ycor[28];
  __shared__ float axw[28], ayw[28];      // mask*0.5*bilinear weights (pool folded)
  int r = blockIdx.x; int b = r / POSTN;
  int tid = threadIdx.x;
  const float* bx = final_boxes + (size_t)r*4;
  float x1 = bx[0], y1 = bx[1], x2 = bx[2], y2 = bx[3];
  float area = (x2-x1)*(y2-y1);
  float lf = floorf(4.0f + log2f(sqrtf(area)/224.0f + 1e-6f));
  lf = fminf(fmaxf(lf, 2.0f), 5.0f);
  int lvl = (int)lf - 2;
  int HW = c_fhw[lvl]; float scale = c_scale[lvl];
  if (tid < 14){
    int i = tid;
    float xs1 = x1*scale, xs2 = x2*scale;
    float bw = fmaxf(xs2-xs1, 1.0f) * (1.0f/7.0f);
    float g = ((float)i + 0.5f)*0.5f;
    float xv = xs1 + bw*g;
    int vx = (xv >= -1.0f) && (xv <= (float)HW);
    float xc = fminf(fmaxf(xv, 0.0f), (float)(HW-1));
    int xl = (int)floorf(xc);
    int xh = min(xl+1, HW-1);
    float LX = xc - (float)xl;
    xcor[2*i]=xl; xcor[2*i+1]=xh;
    axw[2*i]   = 0.5f*(float)vx*(1.0f-LX);
    axw[2*i+1] = 0.5f*(float)vx*LX;
  } else if (tid < 28){
    int i = tid-14;
    float ys1 = y1*scale, ys2 = y2*scale;
    float bh = fmaxf(ys2-ys1, 1.0f)*(1.0f/7.0f);
    float g = ((float)i + 0.5f)*0.5f;
    float yv = ys1 + bh*g;
    int vy = (yv >= -1.0f) && (yv <= (float)HW);
    float yc = fminf(fmaxf(yv, 0.0f), (float)(HW-1));
    int yl = (int)floorf(yc);
    int yh = min(yl+1, HW-1);
    float LY = yc - (float)yl;
    ycor[2*i]=yl; ycor[2*i+1]=yh;
    ayw[2*i]   = 0.5f*(float)vy*(1.0f-LY);
    ayw[2*i+1] = 0.5f*(float)vy*LY;
  }
  __syncthreads();
  for (int p=tid; p<784; p+=128){        // per-ROI weight+owner table, reused
    int t = p / 28, u = p - t*28;        // across all 16 channel tiles
    awp[p][0] = ayw[t]*axw[u];
    awp[p][1] = (float)((t>>2)*7 + (u>>2));
  }
  __syncthreads();
  const float* f  = fp.f[lvl];
  const float* ft = fp.ft[lvl];
  int lane = tid & 31, wave = tid >> 5;
  int n = lane & 15;                       // channel within tile (B/C column)
  // A 16x4 layout (ISA 7.12.2): lane L: M=L%16, VGPR0 holds K=(L<16?0:2), VGPR1 K+1
  float cellf = (float)((wave<<4) + (lane & 15));  // padded rows >=49 never match owner
  int khoff = (lane < 16) ? 0 : 2;
  for (int ct=0; ct<16; ++ct){             // 16 channel tiles of 16 = 256
    for (int e=tid; e<784*16; e+=128){     // stage G into LDS
      int p = e >> 4, cc = e & 15;
      int t = p / 28, u = p - t*28;
      int yy = ycor[t], xx = xcor[u];
      int ch = (ct<<4) + cc;
      float v;
      if (use_t) v = ft[((size_t)(b*HW + yy)*HW + xx)*256 + ch];
      else       v = f[(((size_t)(b*256 + ch))*HW + yy)*HW + xx];
      Gs[p][cc] = v;
    }
    __syncthreads();
    v8f acc = {0,0,0,0,0,0,0,0};
#if HAVE_WMMA
    for (int ks=0; ks<196; ++ks){          // K = 784 in steps of 4
      int kh = (ks<<2) + khoff;            // kh even -> awp[kh] is 16B-aligned
      float w0 = awp[kh][0],   o0 = awp[kh][1];
      float w1 = awp[kh+1][0], o1 = awp[kh+1][1];
      float a0 = (o0 == cellf) ? w0 : 0.0f;
      float a1 = (o1 == cellf) ? w1 : 0.0f;
      v2f av = {a0, a1};
      v2f bv = {Gs[kh][n], Gs[kh+1][n]};   // B 4x16 mirrored to A's K interleave
      acc = __builtin_amdgcn_wmma_f32_16x16x4_f32(
              false, av, false, bv, (short)0, acc, false, false);
    }
#else
    // sparse fallback: only the owned 4x4 weight block is nonzero per cell
    for (int rr=0; rr<8; ++rr){
      int cl = (wave<<4) + rr + ((lane<16)?0:8);
      if (cl < 49){
        int pyy = cl/7, pxx = cl - pyy*7;
        float s = 0.f;
        for (int t=4*pyy; t<4*pyy+4; ++t)
          for (int u=4*pxx; u<4*pxx+4; ++u)
            s += awp[t*28+u][0]*Gs[t*28+u][n];
        acc[rr] = s;
      }
    }
#endif
    // C/D layout: VGPR rr -> M=rr (lanes 0-15) or rr+8 (lanes 16-31), N=lane%16
    int ch = (ct<<4) + n;
    int mbase = (wave<<4) + ((lane<16)?0:8);
    float* OB = out_feats + ((size_t)r*256 + ch)*49;
    for (int rr=0; rr<8; ++rr){
      int cl = mbase + rr;
      if (cl < 49) OB[cl] = acc[rr];
    }
    __syncthreads();
  }
}

// ------------------------------------------------------------------ launcher
extern "C" void kernel_launch(void* const* d_in, const int* in_sizes, int n_in,
                              void* d_out, int out_size, void* d_ws, size_t ws_size,
                              hipStream_t stream){
  (void)n_in; (void)out_size;
  int B = in_sizes[1] / 120000;            // obj0 = B*3*200*200
  InPtrs ip;
  for (int l=0;l<5;l++){ ip.obj[l] = (const float*)d_in[1 + 2*l];
                         ip.del[l] = (const float*)d_in[2 + 2*l]; }
  const float* fIn[4];
  for (int l=0;l<4;l++) fIn[l] = (const float*)d_in[11+l];

  char* w = (char*)d_ws; size_t off = 0;
  auto alloc = [&](size_t bytes)->void*{
    off = (off + 255) & ~(size_t)255; void* p = w + off; off += bytes; return p;
  };
  float*    props = (float*)   alloc((size_t)B*NANCH*4*sizeof(float));
  float*    scr   = (float*)   alloc((size_t)B*NANCH*sizeof(float));
  unsigned* thrT  = (unsigned*)alloc((size_t)B*5*sizeof(unsigned));
  int*      needE = (int*)     alloc((size_t)B*5*sizeof(int));
  int*      sel   = (int*)     alloc((size_t)B*NSEL*sizeof(int));
  unsigned long long* keys = (unsigned long long*)alloc((size_t)B*KEYPAD*8);
  float*    bsel  = (float*)   alloc((size_t)B*NSEL*4*sizeof(float));
  int*      vsel  = (int*)     alloc((size_t)B*NSEL*sizeof(int));
  float*    maxc  = (float*)   alloc((size_t)B*sizeof(float));
  int*      order = (int*)     alloc((size_t)B*NSEL*sizeof(int));
  size_t base_end = (off + 255) & ~(size_t)255;
  int hw4[4] = {200,100,50,25};
  size_t featT_elems = 0;
  for (int l=0;l<4;l++) featT_elems += (size_t)B*256*hw4[l]*hw4[l];
  int use_t = (base_end + featT_elems*sizeof(float)) <= ws_size;
  float* ftbase[4] = {nullptr,nullptr,nullptr,nullptr};
  if (use_t){
    float* ftb = (float*)(w + base_end);
    size_t o2 = 0;
    for (int l=0;l<4;l++){ ftbase[l] = ftb + o2; o2 += (size_t)B*256*hw4[l]*hw4[l]; }
  }
  float* out = (float*)d_out;
  float* outFeats = out + (size_t)B*POSTN*4;

  int total = B*NANCH;
  k_decode <<<(total+255)/256, 256, 0, stream>>>(ip, props, scr, maxc, B);
  k_thresh <<<B*5, 256, 0, stream>>>(scr, thrT, needE, B);
  k_compact<<<B*5, 256, 0, stream>>>(scr, thrT, needE, sel, B);
  k_pack   <<<B*8, 1024, 0, stream>>>(props, scr, sel, bsel, vsel, keys, maxc, B);
  k_sort   <<<B,   1024, 0, stream>>>(keys, order, B);
  k_nms    <<<B,   1024, 0, stream>>>(bsel, vsel, order, maxc, out, B);
  if (use_t){
    for (int l=0;l<4;l++){
      int tt = B*256*hw4[l]*hw4[l];
      k_transpose<<<(tt+255)/256, 256, 0, stream>>>(fIn[l], ftbase[l], B, hw4[l], hw4[l]);
    }
  }
  FeatPtrs fp;
  for (int l=0;l<4;l++){ fp.f[l]=fIn[l]; fp.ft[l]=ftbase[l]; }
  k_roialign<<<B*POSTN, 128, 0, stream>>>(fp, out, outFeats, B, use_t);
}